// S4Layer_22153441313126
// MI455X (gfx1250) — compile-verified
//
#include <hip/hip_runtime.h>
#include <hip/hip_bf16.h>
#include <math.h>

typedef __attribute__((ext_vector_type(2))) float v2f;
typedef __attribute__((ext_vector_type(8))) float v8f;
typedef __attribute__((ext_vector_type(2))) int   v2i;
typedef __attribute__((ext_vector_type(4))) int   v4i;

#define S4_L   262144
#define S4_2L  524288
#define S4_N1  512      /* stage-1 FFT length (over 'a', stride N2) */
#define S4_N2  1024     /* stage-2 FFT length (over 'b', contiguous) */
#define PI_F   3.14159265358979323846f

#define AS1 __attribute__((address_space(1)))
#define AS3 __attribute__((address_space(3)))

#if defined(__has_builtin)
# if __has_builtin(__builtin_amdgcn_global_load_async_to_lds_b32) && \
     __has_builtin(__builtin_amdgcn_global_load_async_to_lds_b64) && \
     __has_builtin(__builtin_amdgcn_global_load_async_to_lds_b128) && \
     __has_builtin(__builtin_amdgcn_s_wait_asynccnt)
#  define S4_ASYNC 1
# endif
#endif
#ifndef S4_ASYNC
# define S4_ASYNC 0
#endif

__device__ __forceinline__ float2 cmulf(float2 a, float2 b) {
    return make_float2(a.x * b.x - a.y * b.y, a.x * b.y + a.y * b.x);
}

// ---------------------------------------------------------------------------
// Kernel 1: DPLR transfer function at all 2L roots of unity.
// Per 16-frequency tile: real matmul [r_re | r_im] (16x128) @ B (128x8) on
// V_WMMA_F32_16X16X4_F32 (32 chained WMMAs), where B = [[v_re, v_im],
// [-v_im, v_re]] is staged once in LDS.  Then Kf = c*(k00 - k01*k10/(1+k11)).
// ---------------------------------------------------------------------------
__global__ void __launch_bounds__(256)
s4_transfer_wmma(const float* __restrict__ Lre, const float* __restrict__ Lim,
                 const float* __restrict__ Pre, const float* __restrict__ Pim,
                 const float* __restrict__ Bre, const float* __restrict__ Bim,
                 const float* __restrict__ Cri, const float* __restrict__ log_step,
                 float2* __restrict__ Kf)
{
    __shared__ float sh_lre[64];
    __shared__ float sh_lim[64];
    __shared__ float sh_B[128][16];          // logical 128x16 B matrix (cols 8..15 zero)
    __shared__ float sh_out[8][16][8];

    const int tid = threadIdx.x;

    // Zero-fill B (128*16 = 2048 entries, 8 per thread), then fill live columns.
    for (int i = tid; i < 128 * 16; i += 256)
        (&sh_B[0][0])[i] = 0.0f;
    __syncthreads();
    if (tid < 64) {
        const int n = tid;
        float lr = Lre[n], li = Lim[n];
        float pr = Pre[n], pi = Pim[n];
        float br = Bre[n], bi = Bim[n];
        float cr = Cri[2 * n], ci = Cri[2 * n + 1];
        sh_lre[n] = lr; sh_lim[n] = li;
        // v0 = conj(C)*B, v1 = conj(C)*P, v2 = conj(P)*B, v3 = conj(P)*P
        float v0r = cr * br + ci * bi, v0i = cr * bi - ci * br;
        float v1r = cr * pr + ci * pi, v1i = cr * pi - ci * pr;
        float v2r = pr * br + pi * bi, v2i = pr * bi - pi * br;
        float v3r = pr * pr + pi * pi, v3i = 0.0f;
        // rows 0..63 multiply r_re:  Re-out cols get v_re, Im-out cols get v_im
        sh_B[n][0] = v0r; sh_B[n][1] = v1r; sh_B[n][2] = v2r; sh_B[n][3] = v3r;
        sh_B[n][4] = v0i; sh_B[n][5] = v1i; sh_B[n][6] = v2i; sh_B[n][7] = v3i;
        // rows 64..127 multiply r_im: Re-out cols get -v_im, Im-out cols get v_re
        sh_B[64 + n][0] = -v0i; sh_B[64 + n][1] = -v1i;
        sh_B[64 + n][2] = -v2i; sh_B[64 + n][3] = -v3i;
        sh_B[64 + n][4] =  v0r; sh_B[64 + n][5] =  v1r;
        sh_B[64 + n][6] =  v2r; sh_B[64 + n][7] =  v3r;
    }
    __syncthreads();

    const float step = __expf(log_step[0]);
    const float two_over_step = 2.0f / step;

    const int w     = tid >> 5;
    const int lane  = tid & 31;
    const int mrow  = lane & 15;     // frequency row within tile / B column
    const int khalf = lane >> 4;     // which K pair this lane holds
    const int tile  = blockIdx.x * 8 + w;
    const int kf    = tile * 16 + mrow;

    // Omega = e^{-i phi}, phi = pi*k/L  ->  g = (2/step)*i*tan(phi/2),
    // c = 1 + i*tan(phi/2), with phi/2 = pi*k/(2L).
    const float ang = (float)kf * (PI_F / (float)S4_2L);
    const float T   = __tanf(ang);
    const float gim = two_over_step * T;

    v8f acc = {};
    float aim0[16], aim1[16];

    // First half: A columns 0..63 = Re(r).  Stash Im(r) for the second half.
    #pragma unroll
    for (int chunk = 0; chunk < 16; ++chunk) {
        const int n0 = chunk * 4 + khalf * 2;
        const int n1 = n0 + 1;
        float dre0 = -sh_lre[n0], dim0 = gim - sh_lim[n0];
        float dre1 = -sh_lre[n1], dim1 = gim - sh_lim[n1];
        float inv0 = __builtin_amdgcn_rcpf(dre0 * dre0 + dim0 * dim0);
        float inv1 = __builtin_amdgcn_rcpf(dre1 * dre1 + dim1 * dim1);
        v2f a, b;
        a.x = dre0 * inv0;  a.y = dre1 * inv1;
        aim0[chunk] = -dim0 * inv0;
        aim1[chunk] = -dim1 * inv1;
        b.x = sh_B[n0][mrow];
        b.y = sh_B[n1][mrow];
        acc = __builtin_amdgcn_wmma_f32_16x16x4_f32(
                  false, a, false, b, (short)0, acc, false, false);
    }
    // Second half: A columns 64..127 = Im(r).
    #pragma unroll
    for (int chunk = 0; chunk < 16; ++chunk) {
        const int n0 = chunk * 4 + khalf * 2;
        v2f a, b;
        a.x = aim0[chunk];  a.y = aim1[chunk];
        b.x = sh_B[64 + n0][mrow];
        b.y = sh_B[64 + n0 + 1][mrow];
        acc = __builtin_amdgcn_wmma_f32_16x16x4_f32(
                  false, a, false, b, (short)0, acc, false, false);
    }

    // D layout: acc[i] holds row M = i + 8*khalf, column N = mrow.
    if (mrow < 8) {
        #pragma unroll
        for (int i = 0; i < 8; ++i)
            sh_out[w][khalf * 8 + i][mrow] = acc[i];
    }
    __syncthreads();

    if (lane < 16) {
        const int m = lane;
        float k00r = sh_out[w][m][0], k01r = sh_out[w][m][1];
        float k10r = sh_out[w][m][2], k11r = sh_out[w][m][3];
        float k00i = sh_out[w][m][4], k01i = sh_out[w][m][5];
        float k10i = sh_out[w][m][6], k11i = sh_out[w][m][7];
        // q = k01*k10 / (1 + k11)
        float dr = 1.0f + k11r, di = k11i;
        float dinv = __builtin_amdgcn_rcpf(dr * dr + di * di);
        float tr = k01r * k10r - k01i * k10i;
        float ti = k01r * k10i + k01i * k10r;
        float qr = (tr * dr + ti * di) * dinv;
        float qi = (ti * dr - tr * di) * dinv;
        float ar = k00r - qr, ai = k00i - qi;
        // Kf = (1 + i*T) * (k00 - q)   (this lane's T matches its row m)
        Kf[tile * 16 + m] = make_float2(ar - T * ai, ai + T * ar);
    }
}

// ---------------------------------------------------------------------------
// In-LDS forward Stockham radix-2 FFT (auto-sorting, ping-pong buffers).
// ---------------------------------------------------------------------------
__device__ float2* fft_lds(float2* buf0, float2* buf1, int n, int tid, int nthr)
{
    float2* src = buf0;
    float2* dst = buf1;
    int l = n >> 1, s = 1, sh = 0;
    float fl = -PI_F / (float)l;          // angle scale: -pi/l, doubles per stage
    while (l >= 1) {
        for (int bi = tid; bi < (n >> 1); bi += nthr) {
            int q = bi & (s - 1);
            int p = bi >> sh;
            float2 a = src[q + s * p];
            float2 b = src[q + s * (p + l)];
            float2 sum = make_float2(a.x + b.x, a.y + b.y);
            float2 dif = make_float2(a.x - b.x, a.y - b.y);
            float sw, cw;
            __sincosf(fl * (float)p, &sw, &cw);   // e^{-2pi i p/(2l)}
            dst[q + s * (2 * p)]     = sum;
            dst[q + s * (2 * p + 1)] = cmulf(dif, make_float2(cw, sw));
        }
        __syncthreads();
        float2* t = src; src = dst; dst = t;
        l >>= 1; s <<= 1; ++sh; fl *= 2.0f;
    }
    return src;
}

// ---------------------------------------------------------------------------
// Four-step FFT of N = N1*N2 = 2^19.  Stage 1: for each b, FFT_N1 over
// x[N2*a + b], apply W_N^{b*c}, store at [c*N2 + b].  Stage 2: for each c,
// FFT_N2 over contiguous row, scatter X[N1*d + c].
// Input staging uses GLOBAL_LOAD_ASYNC_TO_LDS (ASYNCcnt) when available.
// ---------------------------------------------------------------------------
__global__ void __launch_bounds__(256)
s4_fft_stage1(const float* __restrict__ u_real, const float2* __restrict__ src_c,
              int is_real, float2* __restrict__ out)
{
    __shared__ float2 lb[2][S4_N1];
    const int b = blockIdx.x;        // 0..N2-1
    const int tid = threadIdx.x;
#if S4_ASYNC
    if (is_real) {
        // zero-pad boundary is exactly a == 256 (L/N2 == 256)
        for (int a = tid; a < S4_N1; a += 256) {
            lb[0][a].y = 0.0f;
            if (a >= 256) lb[0][a].x = 0.0f;
        }
        {
            const int a = tid;
            __builtin_amdgcn_global_load_async_to_lds_b32(
                (AS1 int*)(u_real + (a * S4_N2 + b)),
                (AS3 int*)&lb[0][a].x, 0, 0);
        }
    } else {
        for (int a = tid; a < S4_N1; a += 256) {
            __builtin_amdgcn_global_load_async_to_lds_b64(
                (AS1 v2i*)(src_c + (a * S4_N2 + b)),
                (AS3 v2i*)&lb[0][a], 0, 0);
        }
    }
    __builtin_amdgcn_s_wait_asynccnt(0);
#else
    for (int a = tid; a < S4_N1; a += 256) {
        const int n = a * S4_N2 + b;
        float2 v;
        if (is_real) { v.x = (n < S4_L) ? u_real[n] : 0.0f; v.y = 0.0f; }
        else         { v = src_c[n]; }
        lb[0][a] = v;
    }
#endif
    __syncthreads();
    float2* res = fft_lds(lb[0], lb[1], S4_N1, tid, 256);
    for (int c = tid; c < S4_N1; c += 256) {
        float sw, cw;
        __sincosf(-2.0f * PI_F * (float)(b * c) / (float)S4_2L, &sw, &cw);
        out[c * S4_N2 + b] = cmulf(res[c], make_float2(cw, sw));
    }
}

__global__ void __launch_bounds__(256)
s4_fft_stage2(const float2* __restrict__ in, float2* __restrict__ out)
{
    __shared__ float2 lb[2][S4_N2];
    const int c = blockIdx.x;        // 0..N1-1
    const int tid = threadIdx.x;
#if S4_ASYNC
    for (int b = tid * 2; b < S4_N2; b += 512) {      // 16B per async op
        __builtin_amdgcn_global_load_async_to_lds_b128(
            (AS1 v4i*)(in + (c * S4_N2 + b)),
            (AS3 v4i*)&lb[0][b], 0, 0);
    }
    __builtin_amdgcn_s_wait_asynccnt(0);
#else
    for (int b = tid; b < S4_N2; b += 256)
        lb[0][b] = in[c * S4_N2 + b];
#endif
    __syncthreads();
    float2* res = fft_lds(lb[0], lb[1], S4_N2, tid, 256);
    for (int d = tid; d < S4_N2; d += 256)
        out[d * S4_N1 + c] = res[d];
}

// Z[k] = conj(Uf[k] * Kf[k]) / 2L   (conj trick: ifft via forward FFT)
__global__ void __launch_bounds__(256)
s4_pointwise(const float2* __restrict__ Uf, const float2* __restrict__ Kf,
             float2* __restrict__ Z)
{
    const int i = blockIdx.x * 256 + threadIdx.x;
    if (i < S4_2L) {
        float2 p = cmulf(Uf[i], Kf[i]);
        const float s = 1.0f / (float)S4_2L;
        Z[i] = make_float2(p.x * s, -p.y * s);
    }
}

// out[t] = Re(ifft)[t] + D*u[t]  (Re(conj(W)) == W.x)
__global__ void __launch_bounds__(256)
s4_combine(const float2* __restrict__ W, const float* __restrict__ u,
           const float* __restrict__ D, float* __restrict__ out)
{
    const int i = blockIdx.x * 256 + threadIdx.x;
    if (i < S4_L) out[i] = W[i].x + D[0] * u[i];
}

extern "C" void kernel_launch(void* const* d_in, const int* in_sizes, int n_in,
                              void* d_out, int out_size, void* d_ws, size_t ws_size,
                              hipStream_t stream)
{
    const float* u   = (const float*)d_in[0];
    const float* Lre = (const float*)d_in[1];
    const float* Lim = (const float*)d_in[2];
    const float* Pre = (const float*)d_in[3];
    const float* Pim = (const float*)d_in[4];
    const float* Bre = (const float*)d_in[5];
    const float* Bim = (const float*)d_in[6];
    const float* Cri = (const float*)d_in[7];
    const float* D   = (const float*)d_in[8];
    const float* lst = (const float*)d_in[9];
    float* out = (float*)d_out;

    float2* bufA = (float2*)d_ws;          // 2^19 complex
    float2* bufB = bufA + S4_2L;           // 2^19 complex
    float2* Kf   = bufB + S4_2L;           // 2^19 complex  (12 MB total)

    // 1) Kf[k] = transfer function at 2L-th roots of unity (WMMA modal matmul)
    s4_transfer_wmma<<<(S4_2L / 16) / 8, 256, 0, stream>>>(
        Lre, Lim, Pre, Pim, Bre, Bim, Cri, lst, Kf);

    // 2-3) Uf = FFT_{2L}(pad(u))
    s4_fft_stage1<<<S4_N2, 256, 0, stream>>>(u, nullptr, 1, bufA);
    s4_fft_stage2<<<S4_N1, 256, 0, stream>>>(bufA, bufB);          // Uf in bufB

    // 4) Z = conj(Uf*Kf)/2L
    s4_pointwise<<<S4_2L / 256, 256, 0, stream>>>(bufB, Kf, bufA);

    // 5-6) W = FFT_{2L}(Z)  ->  ifft(Uf*Kf) = conj(W)
    s4_fft_stage1<<<S4_N2, 256, 0, stream>>>(nullptr, bufA, 0, bufB);
    s4_fft_stage2<<<S4_N1, 256, 0, stream>>>(bufB, bufA);          // W in bufA

    // 7) out = Re + D*u
    s4_combine<<<S4_L / 256, 256, 0, stream>>>(bufA, u, D, out);
}